// PAGTNLayer_16750372454646
// MI455X (gfx1250) — compile-verified
//
#include <hip/hip_runtime.h>
#include <cstdint>
#include <cstddef>

#define NV   30000       // nodes
#define NE   500000      // edges
#define NH   4           // heads
#define NF   32          // feature dim (in = out = edge)
#define NEG_SLOPE 0.2f

typedef float v2f __attribute__((ext_vector_type(2)));
typedef float v8f __attribute__((ext_vector_type(8)));

__device__ __forceinline__ float leaky(float x) { return x > 0.f ? x : NEG_SLOPE * x; }

// monotone float <-> uint32 encoding so we can use atomicMax(u32) as float max
__device__ __forceinline__ unsigned f2ord(float f) {
    unsigned u = __float_as_uint(f);
    return (u & 0x80000000u) ? ~u : (u | 0x80000000u);
}
__device__ __forceinline__ float ord2f(unsigned u) {
    return (u & 0x80000000u) ? __uint_as_float(u & 0x7FFFFFFFu) : __uint_as_float(~u);
}

// ---- WMMA helpers: D(16x16) = A(16x32) * W[:, n0:n0+16] + bias, fp32 exact ----
// A-fragment (f32 16x4): lane l: m=l&15, VGPRs hold K = 4*kk + 2*(l>>4) + {0,1}
__device__ __forceinline__ void load_afrag(const float* __restrict__ rowp, int lane, v2f a[8]) {
    int kh = (lane >> 4) << 1;
#pragma unroll
    for (int kk = 0; kk < 8; ++kk) {
        a[kk].x = rowp[kk * 4 + kh + 0];
        a[kk].y = rowp[kk * 4 + kh + 1];
    }
}

__device__ __forceinline__ v8f gemm_tile(const v2f a[8], const float* __restrict__ W,
                                         const float* __restrict__ bias, int n0, int lane) {
    int n  = (lane & 15) + n0;
    int kh = (lane >> 4) << 1;
    v8f c = {0.f, 0.f, 0.f, 0.f, 0.f, 0.f, 0.f, 0.f};
#pragma unroll
    for (int kk = 0; kk < 8; ++kk) {
        int k = kk * 4 + kh;
        v2f b;
        b.x = W[(k + 0) * NF + n];   // W is (32 x 32) row-major (K x N)
        b.y = W[(k + 1) * NF + n];
        c = __builtin_amdgcn_wmma_f32_16x16x4_f32(false, a[kk], false, b,
                                                  (short)0, c, false, false);
    }
    float bv = bias[n];
#pragma unroll
    for (int r = 0; r < 8; ++r) c[r] += bv;
    return c;
}

// store C/D fragment (ISA layout: lanes 0-15 -> M=r, lanes 16-31 -> M=8+r, N=lane&15)
__device__ __forceinline__ void store_ctile(v8f c, float* __restrict__ out,
                                            int row0, int n0, int ldo, int lane) {
    int n  = (lane & 15) + n0;
    int mb = (lane >> 4) * 8;
#pragma unroll
    for (int r = 0; r < 8; ++r) out[(size_t)(row0 + mb + r) * ldo + n] = c[r];
}

// ---------------- kernel 0: init scratch ----------------
__global__ void k_init(float* feat, float* ssum, unsigned* smax) {
    size_t i = (size_t)blockIdx.x * blockDim.x + threadIdx.x;
    if (i < (size_t)NV * NH * NF) feat[i] = 0.f;
    if (i < (size_t)NV * NH) {
        ssum[i] = 0.f;
        smax[i] = f2ord(-3.402823466e38f);
    }
}

// ---------------- kernel 1: five node projections via WMMA ----------------
__global__ __launch_bounds__(32) void k_node_proj(
    const float* __restrict__ nf,
    const float* W_as, const float* b_as, const float* W_ad, const float* b_ad,
    const float* W_ms, const float* b_ms, const float* W_md, const float* b_md,
    const float* W_wn, const float* b_wn,
    float* src_p, float* dst_p, float* ms_p, float* md_p, float* wn_p) {
    int lane = threadIdx.x;
    int row0 = blockIdx.x * 16;                 // rows over V*H = 120000 (7500 tiles)
    const float* rowp = nf + (size_t)(row0 + (lane & 15)) * NF;
    v2f a[8];
    load_afrag(rowp, lane, a);
#pragma unroll
    for (int nt = 0; nt < 2; ++nt) {
        store_ctile(gemm_tile(a, W_as, b_as, nt * 16, lane), src_p, row0, nt * 16, NF, lane);
        store_ctile(gemm_tile(a, W_ad, b_ad, nt * 16, lane), dst_p, row0, nt * 16, NF, lane);
        store_ctile(gemm_tile(a, W_ms, b_ms, nt * 16, lane), ms_p,  row0, nt * 16, NF, lane);
        store_ctile(gemm_tile(a, W_md, b_md, nt * 16, lane), md_p,  row0, nt * 16, NF, lane);
        store_ctile(gemm_tile(a, W_wn, b_wn, nt * 16, lane), wn_p,  row0, nt * 16, NF, lane);
    }
}

// ---------------- kernel 2: edge attention scores + segment max ----------------
__global__ __launch_bounds__(32) void k_edge_scores(
    const float* __restrict__ ef, const int* __restrict__ src_idx, const int* __restrict__ dst_idx,
    const float* W_ae, const float* b_ae, const float* W_dot, const float* b_dot,
    const float* __restrict__ src_p, const float* __restrict__ dst_p,
    float* __restrict__ scores, unsigned* __restrict__ smax) {
    __shared__ float tile[16 * NF];             // edg_atn tile, 2 KB
    int lane = threadIdx.x;
    int e0 = blockIdx.x * 16;                   // 31250 tiles exactly
    v2f a[8];
    load_afrag(ef + (size_t)(e0 + (lane & 15)) * NF, lane, a);
#pragma unroll
    for (int nt = 0; nt < 2; ++nt)
        store_ctile(gemm_tile(a, W_ae, b_ae, nt * 16, lane), tile, 0, nt * 16, NF, lane);
    __syncthreads();
    // 16 edges x 4 heads = 64 pairs, 32 lanes x 2 iterations
#pragma unroll
    for (int it = 0; it < 2; ++it) {
        int p  = lane + 32 * it;
        int le = p & 15, h = p >> 4;
        int e  = e0 + le;
        int s  = src_idx[e], d = dst_idx[e];
        const float* sp = src_p + ((size_t)s * NH + h) * NF;
        const float* dp = dst_p + ((size_t)d * NH + h) * NF;
        const float* ep = tile + le * NF;
        float sc = b_dot[0];
#pragma unroll
        for (int j = 0; j < NF; ++j)
            sc += leaky(sp[j] + dp[j] + ep[j]) * W_dot[j];
        scores[(size_t)e * NH + h] = sc;
        atomicMax(&smax[(size_t)d * NH + h], f2ord(sc));
    }
}

// ---------------- kernel 3: exp + segment sum ----------------
__global__ void k_exp_sum(float* __restrict__ sc, const int* __restrict__ dst_idx,
                          const unsigned* __restrict__ smax, float* __restrict__ ssum) {
    size_t i = (size_t)blockIdx.x * blockDim.x + threadIdx.x;
    if (i >= (size_t)NE * NH) return;
    int e = (int)(i >> 2), h = (int)(i & 3);
    int d = dst_idx[e];
    float ex = expf(sc[i] - ord2f(smax[(size_t)d * NH + h]));
    sc[i] = ex;
    atomicAdd(&ssum[(size_t)d * NH + h], ex);
}

// ---------------- kernel 4: messages + alpha-weighted scatter add ----------------
__global__ __launch_bounds__(32) void k_message(
    const float* __restrict__ ef, const int* __restrict__ src_idx, const int* __restrict__ dst_idx,
    const float* W_me, const float* b_me,
    const float* __restrict__ ms_p, const float* __restrict__ md_p,
    const float* __restrict__ ex, const float* __restrict__ ssum,
    float* __restrict__ feat) {
    __shared__ float tile[16 * NF];             // msg_e tile
    int lane = threadIdx.x;
    int e0 = blockIdx.x * 16;
    v2f a[8];
    load_afrag(ef + (size_t)(e0 + (lane & 15)) * NF, lane, a);
#pragma unroll
    for (int nt = 0; nt < 2; ++nt)
        store_ctile(gemm_tile(a, W_me, b_me, nt * 16, lane), tile, 0, nt * 16, NF, lane);
    __syncthreads();
#pragma unroll
    for (int it = 0; it < 2; ++it) {
        int p  = lane + 32 * it;
        int le = p & 15, h = p >> 4;
        int e  = e0 + le;
        int s  = src_idx[e], d = dst_idx[e];
        float alpha = ex[(size_t)e * NH + h] / ssum[(size_t)d * NH + h];
        const float* sp = ms_p + ((size_t)s * NH + h) * NF;
        const float* dp = md_p + ((size_t)d * NH + h) * NF;
        const float* ep = tile + le * NF;
        float* fo = feat + ((size_t)d * NH + h) * NF;
#pragma unroll
        for (int j = 0; j < NF; ++j)
            atomicAdd(&fo[j], alpha * leaky(sp[j] + dp[j] + ep[j]));
    }
}

// ---------------- kernel 5: out = act(feat + node@W_wn + b_wn) ----------------
__global__ void k_out(const float* __restrict__ feat, const float* __restrict__ wn_p,
                      float* __restrict__ out) {
    size_t i = (size_t)blockIdx.x * blockDim.x + threadIdx.x;
    if (i >= (size_t)NV * NH * NF) return;
    out[i] = leaky(feat[i] + wn_p[i]);
}

extern "C" void kernel_launch(void* const* d_in, const int* in_sizes, int n_in,
                              void* d_out, int out_size, void* d_ws, size_t ws_size,
                              hipStream_t stream) {
    const float* nf  = (const float*)d_in[0];
    const float* ef  = (const float*)d_in[1];
    const int* src_idx = (const int*)d_in[2];
    const int* dst_idx = (const int*)d_in[3];
    const float* W_as = (const float*)d_in[4],  *b_as = (const float*)d_in[5];
    const float* W_ad = (const float*)d_in[6],  *b_ad = (const float*)d_in[7];
    const float* W_ae = (const float*)d_in[8],  *b_ae = (const float*)d_in[9];
    const float* W_dot= (const float*)d_in[10], *b_dot= (const float*)d_in[11];
    const float* W_ms = (const float*)d_in[12], *b_ms = (const float*)d_in[13];
    const float* W_md = (const float*)d_in[14], *b_md = (const float*)d_in[15];
    const float* W_me = (const float*)d_in[16], *b_me = (const float*)d_in[17];
    const float* W_wn = (const float*)d_in[18], *b_wn = (const float*)d_in[19];
    float* out = (float*)d_out;

    const size_t NP = (size_t)NV * NH * NF;     // 3,840,000
    const size_t NEH = (size_t)NE * NH;         // 2,000,000
    float* ws = (float*)d_ws;
    float*    src_p  = ws;            ws += NP;
    float*    dst_p  = ws;            ws += NP;
    float*    ms_p   = ws;            ws += NP;
    float*    md_p   = ws;            ws += NP;
    float*    wn_p   = ws;            ws += NP;
    float*    feat   = ws;            ws += NP;
    float*    scores = ws;            ws += NEH;
    float*    ssum   = ws;            ws += (size_t)NV * NH;
    unsigned* smax   = (unsigned*)ws;

    const int node_tiles = (NV * NH) / 16;      // 7500
    const int edge_tiles = NE / 16;             // 31250

    k_init<<<(unsigned)((NP + 255) / 256), 256, 0, stream>>>(feat, ssum, smax);
    k_node_proj<<<node_tiles, 32, 0, stream>>>(nf, W_as, b_as, W_ad, b_ad,
                                               W_ms, b_ms, W_md, b_md, W_wn, b_wn,
                                               src_p, dst_p, ms_p, md_p, wn_p);
    k_edge_scores<<<edge_tiles, 32, 0, stream>>>(ef, src_idx, dst_idx, W_ae, b_ae,
                                                 W_dot, b_dot, src_p, dst_p, scores, smax);
    k_exp_sum<<<(unsigned)((NEH + 255) / 256), 256, 0, stream>>>(scores, dst_idx, smax, ssum);
    k_message<<<edge_tiles, 32, 0, stream>>>(ef, src_idx, dst_idx, W_me, b_me,
                                             ms_p, md_p, scores, ssum, feat);
    k_out<<<(unsigned)((NP + 255) / 256), 256, 0, stream>>>(feat, wn_p, out);
}